// S4DLayer_52596169507503
// MI455X (gfx1250) — compile-verified
//
#include <hip/hip_runtime.h>

typedef __attribute__((ext_vector_type(2))) float v2f;
typedef __attribute__((ext_vector_type(8))) float v8f;

#define S4D_H 32              // complex modes
#define S4D_L 16384           // sequence length
#define S4D_B 256             // batch
#define S4D_T 64              // chunk length
#define S4D_C (S4D_L / S4D_T) // 256 chunks
#define S4D_R (S4D_B * S4D_C) // 65536 rows of the chunked view
#define TAB_BYTES 65536       // tables region at start of d_ws

// Pair-interleaved LDS layout for 64x64 B operands:
//   element (row k, col n) -> lds[(k>>1)*PSTR + 2*n + (k&1)]
// so the WMMA B fragment (rows k,k+1 at one column) is ONE aligned ds_load_b64.
// PSTR = 160 dwords == 32 (mod 64 banks): lane halves (pair p vs p+1) hit
// disjoint bank sets 0-31 / 32-63 -> conflict-free b64 reads.
#define PSTR 160
#define PLDS (32 * PSTR)      // dwords per staged matrix (20 KB)

// Tables built on device by s4d_setup:
//  lamT : lambda_h^T (interleaved re,im)
//  V[j][n] : n=2h -> Re(lambda_h^{T-1-j}), n=2h+1 -> Im(...)      (chunk summary)
//  W[k][j] : k=2h -> 2Re(C_h lambda_h^{j+1}), k=2h+1 -> -2Im(...) (carry-in)
//  M[s][j] : upper-triangular Toeplitz of truncated kernel K[j-s] (local conv)
struct S4DTables {
    float lamT[2 * S4D_H];
    float V[S4D_T * 64];
    float W[64 * S4D_T];
    float M[S4D_T * S4D_T];
};

// gfx1250 async copy global->LDS (GVS mode): one b32 per lane, ASYNCcnt-tracked.
__device__ __forceinline__ void s4d_async_ld_b32(unsigned lds_off,
                                                 const void* sbase,
                                                 unsigned byte_off) {
    asm volatile("global_load_async_to_lds_b32 %0, %1, %2"
                 :: "v"(lds_off), "v"(byte_off), "s"(sbase)
                 : "memory");
}
__device__ __forceinline__ void s4d_wait_async0() {
    asm volatile("s_wait_asynccnt 0x0" ::: "memory");
}
// stage one 64x64 row-major table into pair-interleaved LDS layout
__device__ __forceinline__ void s4d_stage_pairswz(float* lds, const float* gsrc) {
    for (int i = threadIdx.x; i < 64 * 64; i += 256) {
        const int row = i >> 6, col = i & 63;
        const unsigned dst = (unsigned)(unsigned long long)
            &lds[(row >> 1) * PSTR + col * 2 + (row & 1)];
        s4d_async_ld_b32(dst, gsrc, (unsigned)(i * 4));
    }
}

__global__ void s4d_setup(const float* __restrict__ log_A_real,
                          const float* __restrict__ A_imag,
                          const float* __restrict__ C_real,
                          const float* __restrict__ C_imag,
                          const float* __restrict__ log_step,
                          S4DTables* __restrict__ tab) {
    __shared__ float pwRe[S4D_H][S4D_T + 1];
    __shared__ float pwIm[S4D_H][S4D_T + 1];
    __shared__ float Ks[S4D_T];
    const int tid = threadIdx.x;          // 64 threads
    if (tid < S4D_H) {
        const int h = tid;
        const float step = __expf(log_step[0]);
        const float dre = -__expf(log_A_real[h]) * step;
        const float dim = A_imag[h] * step;
        const float er = __expf(dre);
        float sn, cs;
        __sincosf(dim, &sn, &cs);
        const float lre = er * cs, lim = er * sn;   // lambda_h = exp(dt*A_h)
        float pre = 1.0f, pim = 0.0f;
        for (int j = 0; j <= S4D_T; ++j) {
            pwRe[h][j] = pre; pwIm[h][j] = pim;
            const float nre = pre * lre - pim * lim;
            const float nim = pre * lim + pim * lre;
            pre = nre; pim = nim;
        }
        tab->lamT[2 * h]     = pwRe[h][S4D_T];
        tab->lamT[2 * h + 1] = pwIm[h][S4D_T];
    }
    __syncthreads();
    {   // thread tid owns index j = tid (0..63)
        const int j = tid;
        float kacc = 0.0f;
        for (int h = 0; h < S4D_H; ++h) {
            const float cr = C_real[h], ci = C_imag[h];
            tab->V[j * 64 + 2 * h]     = pwRe[h][S4D_T - 1 - j];
            tab->V[j * 64 + 2 * h + 1] = pwIm[h][S4D_T - 1 - j];
            const float qr = pwRe[h][j + 1], qi = pwIm[h][j + 1];
            tab->W[(2 * h) * S4D_T + j]     =  2.0f * (cr * qr - ci * qi);
            tab->W[(2 * h + 1) * S4D_T + j] = -2.0f * (cr * qi + ci * qr);
            kacc += 2.0f * (cr * pwRe[h][j] - ci * pwIm[h][j]);
        }
        Ks[j] = kacc;
    }
    __syncthreads();
    {
        const int sp = tid;
        for (int j = 0; j < S4D_T; ++j)
            tab->M[sp * S4D_T + j] = (j >= sp) ? Ks[j - sp] : 0.0f;
    }
}

// ---- GEMM 1: Z (R x 64) = U_chunks (R x 64) @ V (64 x 64), f32 WMMA ----
// One wave owns a 16-row slab and all 4 column tiles (A-fragment reuse x4).
__global__ void __launch_bounds__(256)
s4d_chunk_summary(const float* __restrict__ u, const S4DTables* __restrict__ tab,
                  float* __restrict__ Z) {
    __shared__ float ldsV[PLDS];
    s4d_stage_pairswz(ldsV, tab->V);
    s4d_wait_async0();
    __syncthreads();

    const int lane = threadIdx.x & 31;
    const int wave = threadIdx.x >> 5;
    const int row0 = (blockIdx.x * 8 + wave) * 16;   // 4096 row slabs
    const int m    = lane & 15;
    const int half = lane >> 4;

    const float* aRow = u + (size_t)(row0 + m) * S4D_T;
    __builtin_prefetch(aRow + 128 * S4D_T, 0, 1);    // global_prefetch_b8

    v8f acc[4] = {};
    for (int k0 = 0; k0 < S4D_T; k0 += 4) {
        // A 16x4 f32 fragment: lane holds K = k0+2*half, k0+2*half+1
        const v2f a = *(const v2f*)(aRow + k0 + 2 * half);
        const int p = (k0 >> 1) + half;              // B pair-row index
        #pragma unroll
        for (int nt = 0; nt < 4; ++nt) {
            const v2f b = *(const v2f*)&ldsV[p * PSTR + (nt * 16 + m) * 2];
            acc[nt] = __builtin_amdgcn_wmma_f32_16x16x4_f32(
                false, a, false, b, (short)0, acc[nt], false, false);
        }
    }
    // C/D layout: VGPR r -> row (row0 + 8*half + r), col nt*16 + m
    float* zBase = Z + (size_t)(row0 + half * 8) * 64 + m;
    #pragma unroll
    for (int nt = 0; nt < 4; ++nt)
        #pragma unroll
        for (int r = 0; r < 8; ++r)
            zBase[(size_t)r * 64 + nt * 16] = acc[nt][r];
}

// ---- Inter-chunk linear scan, in place: Z[b,c] becomes state BEFORE chunk c ----
__global__ void s4d_scan(float* __restrict__ Zf, const S4DTables* __restrict__ tab) {
    const int t = blockIdx.x * blockDim.x + threadIdx.x;  // 8192 = B*H
    const int b = t >> 5;
    const int h = t & 31;
    const float lre = tab->lamT[2 * h], lim = tab->lamT[2 * h + 1];
    v2f* Z = (v2f*)Zf;                                    // 32 complex per row
    float sre = 0.0f, sim = 0.0f;
    size_t idx = ((size_t)b * S4D_C) * 32 + h;
    for (int c = 0; c < S4D_C; ++c) {
        const v2f z = Z[idx];
        v2f s; s.x = sre; s.y = sim;
        Z[idx] = s;                                       // pre-chunk state
        const float nre = lre * sre - lim * sim + z.x;
        const float nim = lre * sim + lim * sre + z.y;
        sre = nre; sim = nim;
        idx += 32;
    }
}

// ---- GEMM 2: Y = U@M + S@W + D*U (two accumulating WMMA loops, 4 col tiles) ----
__global__ void __launch_bounds__(256)
s4d_output(const float* __restrict__ u, const float* __restrict__ S,
           const S4DTables* __restrict__ tab, const float* __restrict__ Dptr,
           float* __restrict__ y) {
    __shared__ float ldsM[PLDS];
    __shared__ float ldsW[PLDS];
    s4d_stage_pairswz(ldsM, tab->M);
    s4d_stage_pairswz(ldsW, tab->W);
    s4d_wait_async0();
    __syncthreads();

    const int lane = threadIdx.x & 31;
    const int wave = threadIdx.x >> 5;
    const int row0 = (blockIdx.x * 8 + wave) * 16;
    const int m    = lane & 15;
    const int half = lane >> 4;

    const float* aRow = u + (size_t)(row0 + m) * S4D_T;
    const float* sRow = S + (size_t)(row0 + m) * 64;
    __builtin_prefetch(aRow + 128 * S4D_T, 0, 1);

    v8f acc[4] = {};
    for (int k0 = 0; k0 < S4D_T; k0 += 4) {           // local causal conv: U @ M
        const v2f a = *(const v2f*)(aRow + k0 + 2 * half);
        const int p = (k0 >> 1) + half;
        #pragma unroll
        for (int nt = 0; nt < 4; ++nt) {
            const v2f b = *(const v2f*)&ldsM[p * PSTR + (nt * 16 + m) * 2];
            acc[nt] = __builtin_amdgcn_wmma_f32_16x16x4_f32(
                false, a, false, b, (short)0, acc[nt], false, false);
        }
    }
    for (int k0 = 0; k0 < 64; k0 += 4) {              // carry-in: S @ W
        const v2f a = *(const v2f*)(sRow + k0 + 2 * half);
        const int p = (k0 >> 1) + half;
        #pragma unroll
        for (int nt = 0; nt < 4; ++nt) {
            const v2f b = *(const v2f*)&ldsW[p * PSTR + (nt * 16 + m) * 2];
            acc[nt] = __builtin_amdgcn_wmma_f32_16x16x4_f32(
                false, a, false, b, (short)0, acc[nt], false, false);
        }
    }

    const float Dv = Dptr[0];
    float* yBase = y + (size_t)(row0 + half * 8) * 64 + m;
    const float* uBase = u + (size_t)(row0 + half * 8) * S4D_T + m;
    #pragma unroll
    for (int nt = 0; nt < 4; ++nt)
        #pragma unroll
        for (int r = 0; r < 8; ++r)
            yBase[(size_t)r * 64 + nt * 16] =
                acc[nt][r] + Dv * uBase[(size_t)r * 64 + nt * 16];
}

extern "C" void kernel_launch(void* const* d_in, const int* in_sizes, int n_in,
                              void* d_out, int out_size, void* d_ws, size_t ws_size,
                              hipStream_t stream) {
    const float* u          = (const float*)d_in[0];
    const float* log_A_real = (const float*)d_in[1];
    const float* A_imag     = (const float*)d_in[2];
    const float* C_real     = (const float*)d_in[3];
    const float* C_imag     = (const float*)d_in[4];
    const float* Dp         = (const float*)d_in[5];
    const float* log_step   = (const float*)d_in[6];
    float* y = (float*)d_out;

    char* ws = (char*)d_ws;
    S4DTables* tab = (S4DTables*)ws;                    // ~48.5 KB
    float* Z = (float*)(ws + TAB_BYTES);                // R*64 f32 = 16.8 MB

    const int slabBlocks = (S4D_R / 16) / 8;            // 512 blocks, 8 waves each

    s4d_setup<<<1, 64, 0, stream>>>(log_A_real, A_imag, C_real, C_imag, log_step, tab);
    s4d_chunk_summary<<<slabBlocks, 256, 0, stream>>>(u, tab, Z);
    s4d_scan<<<(S4D_B * S4D_H) / 256, 256, 0, stream>>>(Z, tab);
    s4d_output<<<slabBlocks, 256, 0, stream>>>(u, Z, tab, Dp, y);
}